// ContrastiveLearningWrapper_18794776887725
// MI455X (gfx1250) — compile-verified
//
#include <hip/hip_runtime.h>

typedef __attribute__((ext_vector_type(2))) float v2f;
typedef __attribute__((ext_vector_type(8))) float v8f;

#define N_ROWS 4096
#define D_DIM  512
#define P_DIM  128
#define INV_TEMP 10.0f   // 1 / 0.1
#define NL_W     0.5
#define CLAMP_V  100.0f  // -LOG_CLAMP

// ---------------------------------------------------------------------------
// Kernel 1: z = rownorm(x @ W^T + b)   (4096x512 * 512x128 -> 4096x128)
// One block = 16 rows of x. 8 waves; wave w owns columns [16w, 16w+16).
// K-loop of V_WMMA_F32_16X16X4_F32.
// ---------------------------------------------------------------------------
__global__ void __launch_bounds__(256)
zgemm_norm_kernel(const float* __restrict__ x,
                  const float* __restrict__ W,
                  const float* __restrict__ bias,
                  float* __restrict__ z) {
    __shared__ float zt[16 * P_DIM];   // 16 x 128 tile of z (pre-norm)
    __shared__ float rnorm[16];

    const int tid     = threadIdx.x;
    const int wave    = tid >> 5;
    const int lane    = tid & 31;
    const int r0      = blockIdx.x * 16;
    const int pBase   = wave * 16;
    const int halfsel = lane >> 4;         // 0: lanes 0-15, 1: lanes 16-31
    const int mn      = lane & 15;
    const int kk      = halfsel * 2;       // A/B frag K sub-offset (ISA 7.12.2)

    // A frag: row (r0+mn) of x, K pairs {k+kk, k+kk+1}
    const float* arow = x + (size_t)(r0 + mn) * D_DIM + kk;
    // B frag: B[k][p] = W[p][k]  (W is P x D row-major)
    const float* brow = W + (size_t)(pBase + mn) * D_DIM + kk;

    v8f c = {};
#pragma unroll 8
    for (int k = 0; k < D_DIM; k += 4) {
        v2f a = *(const v2f*)(arow + k);
        v2f b = *(const v2f*)(brow + k);
        c = __builtin_amdgcn_wmma_f32_16x16x4_f32(false, a, false, b,
                                                  (short)0, c, false, false);
    }

    // C/D layout: VGPR v holds (M = v + 8*halfsel, N = mn). Add bias, stash.
    const float bv = bias[pBase + mn];
#pragma unroll
    for (int v = 0; v < 8; ++v) {
        int m = v + halfsel * 8;
        zt[m * P_DIM + pBase + mn] = c[v] + bv;
    }
    __syncthreads();

    if (tid < 16) {
        float s = 0.0f;
        for (int p = 0; p < P_DIM; ++p) {
            float t = zt[tid * P_DIM + p];
            s += t * t;
        }
        rnorm[tid] = rsqrtf(s);
    }
    __syncthreads();

    // 2048 elements / 256 threads = 8 contiguous each (stays in one row)
    {
        int base = tid * 8;
        int row  = base >> 7;              // /128
        float rn = rnorm[row];
        float* dst = z + (size_t)(r0 + row) * P_DIM + (base & 127);
        const float* src = zt + base;
#pragma unroll
        for (int i = 0; i < 8; ++i) dst[i] = src[i] * rn;
    }
}

// ---------------------------------------------------------------------------
// Kernel 2: predictions = sigmoid(z z^T / T), fused masked-BCE reduction.
// One wave owns a 32x64 output strip: 2 A-frags x 4 B-frags -> 8 C tiles.
// Per k-step: 6 global_load_b64 feed 8 v_wmma_f32_16x16x4_f32.
// 8 waves per block share the same 32 rows (A reuse through WGP$).
// ---------------------------------------------------------------------------
__global__ void __launch_bounds__(256)
simloss_kernel(const float* __restrict__ z,
               const int*   __restrict__ tgt,
               float*       __restrict__ pred,     // d_out + 1, row-major NxN
               double*      __restrict__ accum,    // [0]=pos_sum [1]=neg_sum
               unsigned long long* __restrict__ cnt) {
    __shared__ double s_pos, s_neg;
    __shared__ unsigned long long s_cnt;

    const int tid = threadIdx.x;
    if (tid == 0) { s_pos = 0.0; s_neg = 0.0; s_cnt = 0ull; }
    __syncthreads();

    const int wave    = tid >> 5;
    const int lane    = tid & 31;
    const int gw      = blockIdx.x * 8 + wave;   // 8192 strips total
    const int rt      = gw >> 6;                 // 128 row strips (32 rows)
    const int ct      = gw & 63;                 // 64 col strips (64 cols)
    const int halfsel = lane >> 4;
    const int mn      = lane & 15;
    const int kk      = halfsel * 2;

    const int rowBase = rt * 32;
    const int colBase = ct * 64;

    // A frags: rows rowBase + {0,16} + mn ; B frags: cols colBase + 16j + mn
    const float* arow0 = z + (size_t)(rowBase +      mn) * P_DIM + kk;
    const float* arow1 = z + (size_t)(rowBase + 16 + mn) * P_DIM + kk;
    const float* brow0 = z + (size_t)(colBase +      mn) * P_DIM + kk;
    const float* brow1 = z + (size_t)(colBase + 16 + mn) * P_DIM + kk;
    const float* brow2 = z + (size_t)(colBase + 32 + mn) * P_DIM + kk;
    const float* brow3 = z + (size_t)(colBase + 48 + mn) * P_DIM + kk;

    v8f c00 = {}, c01 = {}, c02 = {}, c03 = {};
    v8f c10 = {}, c11 = {}, c12 = {}, c13 = {};

#pragma unroll 2
    for (int k = 0; k < P_DIM; k += 4) {
        v2f a0 = *(const v2f*)(arow0 + k);
        v2f a1 = *(const v2f*)(arow1 + k);
        v2f b0 = *(const v2f*)(brow0 + k);
        v2f b1 = *(const v2f*)(brow1 + k);
        v2f b2 = *(const v2f*)(brow2 + k);
        v2f b3 = *(const v2f*)(brow3 + k);
        c00 = __builtin_amdgcn_wmma_f32_16x16x4_f32(false, a0, false, b0, (short)0, c00, false, false);
        c01 = __builtin_amdgcn_wmma_f32_16x16x4_f32(false, a0, false, b1, (short)0, c01, false, false);
        c02 = __builtin_amdgcn_wmma_f32_16x16x4_f32(false, a0, false, b2, (short)0, c02, false, false);
        c03 = __builtin_amdgcn_wmma_f32_16x16x4_f32(false, a0, false, b3, (short)0, c03, false, false);
        c10 = __builtin_amdgcn_wmma_f32_16x16x4_f32(false, a1, false, b0, (short)0, c10, false, false);
        c11 = __builtin_amdgcn_wmma_f32_16x16x4_f32(false, a1, false, b1, (short)0, c11, false, false);
        c12 = __builtin_amdgcn_wmma_f32_16x16x4_f32(false, a1, false, b2, (short)0, c12, false, false);
        c13 = __builtin_amdgcn_wmma_f32_16x16x4_f32(false, a1, false, b3, (short)0, c13, false, false);
    }

    v8f* ctile[2][4] = { { &c00, &c01, &c02, &c03 },
                         { &c10, &c11, &c12, &c13 } };

    double pos_s = 0.0, neg_s = 0.0;
    unsigned int pos_c = 0;

    int tcol[4];
#pragma unroll
    for (int j = 0; j < 4; ++j) tcol[j] = tgt[colBase + j * 16 + mn];

#pragma unroll
    for (int i = 0; i < 2; ++i) {
#pragma unroll
        for (int v = 0; v < 8; ++v) {
            const int   row  = rowBase + i * 16 + v + halfsel * 8;
            const int   trow = tgt[row];
            float* prow = pred + (size_t)row * N_ROWS;
#pragma unroll
            for (int j = 0; j < 4; ++j) {
                const int col = colBase + j * 16 + mn;
                float s = (*ctile[i][j])[v] * INV_TEMP;
                // prediction = sigmoid(s)
                float p = 1.0f / (1.0f + __expf(-s));
                prow[col] = p;

                if (row != col) {
                    if (trow == tcol[j]) {
                        // -log(sigmoid(s)) = softplus(-s), clamped at 100
                        float sp = ((-s) > 20.0f) ? -s : log1pf(__expf(-s));
                        pos_s += (double)fminf(sp, CLAMP_V);
                        pos_c++;
                    } else {
                        // -log(1-sigmoid(s)) = softplus(s), clamped at 100
                        float sp = (s > 20.0f) ? s : log1pf(__expf(s));
                        neg_s += (double)fminf(sp, CLAMP_V);
                    }
                }
            }
        }
    }

    atomicAdd(&s_pos, pos_s);
    atomicAdd(&s_neg, neg_s);
    atomicAdd(&s_cnt, (unsigned long long)pos_c);
    __syncthreads();
    if (tid == 0) {
        atomicAdd(&accum[0], s_pos);
        atomicAdd(&accum[1], s_neg);
        atomicAdd(cnt, s_cnt);
    }
}

// ---------------------------------------------------------------------------
// Kernel 3: loss = pos_sum/pos_cnt + 0.5 * neg_sum/neg_cnt
// ---------------------------------------------------------------------------
__global__ void finalize_kernel(const double* __restrict__ accum,
                                const unsigned long long* __restrict__ cnt,
                                float* __restrict__ out) {
    double pc    = (double)(*cnt);
    double total = (double)N_ROWS * (double)(N_ROWS - 1);
    double nc    = total - pc;
    double loss  = accum[0] / pc + NL_W * (accum[1] / nc);
    out[0] = (float)loss;
}

extern "C" void kernel_launch(void* const* d_in, const int* in_sizes, int n_in,
                              void* d_out, int out_size, void* d_ws, size_t ws_size,
                              hipStream_t stream) {
    const float* x    = (const float*)d_in[0];   // outputs (4096,1,512)
    const int*   tgt  = (const int*)  d_in[1];   // targets (4096,)
    const float* W    = (const float*)d_in[2];   // (128,512)
    const float* bias = (const float*)d_in[3];   // (128,)
    float* out = (float*)d_out;                  // [loss, predictions...]

    // Workspace layout: z (4096x128 f32 = 2 MB), then accumulators.
    float* z = (float*)d_ws;
    size_t zbytes = (size_t)N_ROWS * P_DIM * sizeof(float);
    double* accum = (double*)((char*)d_ws + zbytes);
    unsigned long long* cnt = (unsigned long long*)(accum + 2);

    hipMemsetAsync(accum, 0,
                   2 * sizeof(double) + sizeof(unsigned long long), stream);

    zgemm_norm_kernel<<<N_ROWS / 16, 256, 0, stream>>>(x, W, bias, z);

    // 128 x 64 = 8192 strips (32x64 each), 8 waves per 256-thread block
    simloss_kernel<<<8192 / 8, 256, 0, stream>>>(z, tgt, out + 1, accum, cnt);

    finalize_kernel<<<1, 1, 0, stream>>>(accum, cnt, out);
}